// GIN_49460843381582
// MI455X (gfx1250) — compile-verified
//
#include <hip/hip_runtime.h>
#include <math.h>

typedef __attribute__((ext_vector_type(2))) float v2f;
typedef __attribute__((ext_vector_type(8))) float v8f;

#define LN_EPS 1e-5f

// ---------------------------------------------------------------------------
// Utility kernels
// ---------------------------------------------------------------------------
__global__ void copy_f4_kernel(const float4* __restrict__ in,
                               float4* __restrict__ out, int n4) {
    int i = blockIdx.x * blockDim.x + threadIdx.x;
    if (i < n4) out[i] = in[i];
}

__global__ void fill_neginf_kernel(float* __restrict__ p, int n) {
    int i = blockIdx.x * blockDim.x + threadIdx.x;
    if (i < n) p[i] = -INFINITY;
}

// One wave per edge: agg[dst] += feat[src]  (feature dim D, 32-lane strided)
__global__ void scatter_add_kernel(const float* __restrict__ feat,
                                   const int* __restrict__ src,
                                   const int* __restrict__ dst,
                                   float* __restrict__ agg,
                                   int nEdges, int D) {
    int gid  = blockIdx.x * blockDim.x + threadIdx.x;
    int e    = gid >> 5;
    int lane = gid & 31;
    if (e >= nEdges) return;
    int s = src[e];
    int d = dst[e];
    const float* fs = feat + (size_t)s * D;
    float*       fd = agg  + (size_t)d * D;
    for (int j = lane; j < D; j += 32)
        atomicAdd(&fd[j], fs[j]);
}

// One wave per node: global max pool via float atomicMax (D = 256)
__global__ void seg_max_kernel(const float* __restrict__ h,
                               const int* __restrict__ batch,
                               float* __restrict__ g, int nNodes) {
    int gid  = blockIdx.x * blockDim.x + threadIdx.x;
    int node = gid >> 5;
    int lane = gid & 31;
    if (node >= nNodes) return;
    int b = batch[node];
    const float* p = h + (size_t)node * 256;
    float*       q = g + (size_t)b * 256;
#pragma unroll
    for (int i = 0; i < 8; ++i) {
        int j = lane + 32 * i;
        atomicMax(&q[j], p[j]);
    }
}

// ---------------------------------------------------------------------------
// WMMA fp32 GEMM: out[M,N] = act(A[M,K] @ W[K,N] + bias[N])
//
// Block = 128 threads (4 waves), computes a 64x64 output tile:
//   wave w -> rows [mBlk*64 + w*16, +16), cols [nBlk*64, +64) as 4 x (16x16)
//   accumulator fragments (independent V_WMMA_F32_16X16X4_F32 chains).
//
// Weights are staged per 128-deep K-slab into LDS with the CDNA5 async
// global->LDS path (GLOBAL_LOAD_ASYNC_TO_LDS_B32, ASYNCcnt, s_wait_asynccnt).
// The per-lane LDS destination address lets us K-pair-swizzle the tile:
//     sB[ ((k>>1)*64 + col)*2 + (k&1) ]
// so each WMMA B fragment (K=k+kOff, k+kOff+1 at one column) is an 8-byte
// contiguous pair -> a single ds_load_b64 straight into the operand VGPR pair.
// Requires: M%64==0, N%64==0, K%128==0. Grid sized exactly -> EXEC all-ones.
// act: 0 = none, 1 = relu, 2 = leaky_relu(0.2)
// ---------------------------------------------------------------------------
#define KSLAB 128

__global__ void wmma_gemm_kernel(const float* __restrict__ A,
                                 const float* __restrict__ W,
                                 const float* __restrict__ bias,
                                 float* __restrict__ out,
                                 int M, int K, int N, int act) {
    __shared__ float sB[KSLAB * 64];   // 32 KB, K-pair swizzled

    const int tid  = threadIdx.x;
    const int lane = tid & 31;
    const int wave = tid >> 5;

    const int nB64  = N >> 6;
    const int mBlk  = blockIdx.x / nB64;
    const int nBlk  = blockIdx.x - mBlk * nB64;
    const int nBase = nBlk << 6;

    // ISA 7.12.2 fp32 fragment layout:
    //   A 16x4 : lane L holds row M=L&15; VGPR0/1 hold K = kOff, kOff+1
    //   B 4x16 : lane L holds col N=L&15; VGPR0/1 hold K = kOff, kOff+1
    const int kOff = (lane >> 4) << 1;   // 0 for lanes 0-15, 2 for lanes 16-31
    const int colL = lane & 15;
    const int rowA = (mBlk << 6) + (wave << 4) + colL;
    const float* aRow = A + (size_t)rowA * K;

    const unsigned ldsBase = (unsigned)(uintptr_t)(&sB[0]);

    v8f c0 = {}, c1 = {}, c2 = {}, c3 = {};

    for (int k0 = 0; k0 < K; k0 += KSLAB) {
        // ---- async stage W[k0:k0+128, nBase:nBase+64] into LDS, swizzled ----
        // 128 rows x 64 cols, one B32 per element; 128 threads x 64 iters.
#pragma unroll 8
        for (int c = tid; c < KSLAB * 64; c += 128) {
            int kk  = c >> 6;               // 0..127 row within slab
            int col = c & 63;               // 0..63
            const float* gp = W + (size_t)(k0 + kk) * N + (nBase + col);
            unsigned swz = (unsigned)((((kk >> 1) << 7) + (col << 1) + (kk & 1)) << 2);
            unsigned ldsOff = ldsBase + swz;
            asm volatile("global_load_async_to_lds_b32 %0, %1, off"
                         :: "v"(ldsOff), "v"(gp)
                         : "memory");
        }
        asm volatile("s_wait_asynccnt 0" ::: "memory");
        __syncthreads();

        // ---- compute on the slab ----
        const float* aSlab = aRow + k0;
#pragma unroll
        for (int k = 0; k < KSLAB; k += 4) {
            if ((k & 63) == 0)
                __builtin_prefetch(&aSlab[k + KSLAB], 0, 1);  // global_prefetch_b8
            v2f a;
            a.x = aSlab[k + kOff];
            a.y = aSlab[k + kOff + 1];
            // fragment pairs are contiguous in swizzled LDS
            const float* frag = &sB[(((k + kOff) >> 1) << 7) + (colL << 1)];
            v2f b0 = *(const v2f*)(frag +  0);   // cols  0..15
            v2f b1 = *(const v2f*)(frag + 32);   // cols 16..31
            v2f b2 = *(const v2f*)(frag + 64);   // cols 32..47
            v2f b3 = *(const v2f*)(frag + 96);   // cols 48..63
            // (neg_a, A, neg_b, B, c_mod, C, reuse_a, reuse_b)
            c0 = __builtin_amdgcn_wmma_f32_16x16x4_f32(false, a, false, b0,
                                                       (short)0, c0, false, false);
            c1 = __builtin_amdgcn_wmma_f32_16x16x4_f32(false, a, false, b1,
                                                       (short)0, c1, false, false);
            c2 = __builtin_amdgcn_wmma_f32_16x16x4_f32(false, a, false, b2,
                                                       (short)0, c2, false, false);
            c3 = __builtin_amdgcn_wmma_f32_16x16x4_f32(false, a, false, b3,
                                                       (short)0, c3, false, false);
        }
        __syncthreads();   // protect sB before next slab overwrites it
    }

    // ---- epilogue: bias + activation + store ----
    const int mFragBase = (mBlk << 6) + (wave << 4) + ((lane >> 4) << 3);
    v8f accs[4] = {c0, c1, c2, c3};
#pragma unroll
    for (int f = 0; f < 4; ++f) {
        int   colG = nBase + (f << 4) + colL;
        float bv   = bias[colG];
#pragma unroll
        for (int v = 0; v < 8; ++v) {
            int   m   = mFragBase + v;
            float val = accs[f][v] + bv;
            if (act == 1)      val = fmaxf(val, 0.0f);
            else if (act == 2) val = (val > 0.0f) ? val : 0.2f * val;
            out[(size_t)m * N + colG] = val;
        }
    }
}

// ---------------------------------------------------------------------------
// LayerNorm(256) + leaky_relu(0.2), in place. One wave per node.
// ---------------------------------------------------------------------------
__global__ void ln_leaky256_kernel(float* __restrict__ h,
                                   const float* __restrict__ gamma,
                                   const float* __restrict__ beta,
                                   int nRows) {
    int gid  = blockIdx.x * blockDim.x + threadIdx.x;
    int row  = gid >> 5;
    int lane = gid & 31;
    if (row >= nRows) return;
    float* p = h + (size_t)row * 256;

    float vals[8];
    float s = 0.0f, s2 = 0.0f;
#pragma unroll
    for (int i = 0; i < 8; ++i) {
        float v = p[lane + 32 * i];
        vals[i] = v;
        s  += v;
        s2 += v * v;
    }
#pragma unroll
    for (int o = 16; o > 0; o >>= 1) {
        s  += __shfl_xor(s,  o, 32);
        s2 += __shfl_xor(s2, o, 32);
    }
    float mean = s * (1.0f / 256.0f);
    float var  = s2 * (1.0f / 256.0f) - mean * mean;
    float inv  = rsqrtf(var + LN_EPS);
#pragma unroll
    for (int i = 0; i < 8; ++i) {
        int   j = lane + 32 * i;
        float y = (vals[i] - mean) * inv * gamma[j] + beta[j];
        p[j] = (y > 0.0f) ? y : 0.2f * y;
    }
}

// ---------------------------------------------------------------------------
// Head: LN(768) + leaky(0.2) + [768x2] matvec + bias + softmax.
// One wave per graph row.
// ---------------------------------------------------------------------------
__global__ void head_kernel(const float* __restrict__ u,
                            const float* __restrict__ lng,
                            const float* __restrict__ lnb,
                            const float* __restrict__ w2,
                            const float* __restrict__ b2,
                            float* __restrict__ out, int nRows) {
    int gid  = blockIdx.x * blockDim.x + threadIdx.x;
    int row  = gid >> 5;
    int lane = gid & 31;
    if (row >= nRows) return;
    const float* p = u + (size_t)row * 768;

    float vals[24];
    float s = 0.0f, s2 = 0.0f;
#pragma unroll
    for (int i = 0; i < 24; ++i) {
        float v = p[lane + 32 * i];
        vals[i] = v;
        s  += v;
        s2 += v * v;
    }
#pragma unroll
    for (int o = 16; o > 0; o >>= 1) {
        s  += __shfl_xor(s,  o, 32);
        s2 += __shfl_xor(s2, o, 32);
    }
    float mean = s * (1.0f / 768.0f);
    float var  = s2 * (1.0f / 768.0f) - mean * mean;
    float inv  = rsqrtf(var + LN_EPS);

    float a0 = 0.0f, a1 = 0.0f;
#pragma unroll
    for (int i = 0; i < 24; ++i) {
        int   j = lane + 32 * i;
        float y = (vals[i] - mean) * inv * lng[j] + lnb[j];
        y = (y > 0.0f) ? y : 0.2f * y;
        a0 += y * w2[2 * j + 0];
        a1 += y * w2[2 * j + 1];
    }
#pragma unroll
    for (int o = 16; o > 0; o >>= 1) {
        a0 += __shfl_xor(a0, o, 32);
        a1 += __shfl_xor(a1, o, 32);
    }
    if (lane == 0) {
        float z0 = a0 + b2[0];
        float z1 = a1 + b2[1];
        float m  = fmaxf(z0, z1);
        float e0 = expf(z0 - m);
        float e1 = expf(z1 - m);
        float d  = 1.0f / (e0 + e1);
        out[(size_t)row * 2 + 0] = e0 * d;
        out[(size_t)row * 2 + 1] = e1 * d;
    }
}

// ---------------------------------------------------------------------------
// Host-side orchestration
// ---------------------------------------------------------------------------
extern "C" void kernel_launch(void* const* d_in, const int* in_sizes, int n_in,
                              void* d_out, int out_size, void* d_ws, size_t ws_size,
                              hipStream_t stream) {
    const float* x       = (const float*)d_in[0];
    const int*   ei      = (const int*)  d_in[1];
    const int*   batch   = (const int*)  d_in[2];
    const float* g1w1    = (const float*)d_in[3];
    const float* g1b1    = (const float*)d_in[4];
    const float* g1w2    = (const float*)d_in[5];
    const float* g1b2    = (const float*)d_in[6];
    const float* g2w1    = (const float*)d_in[7];
    const float* g2b1    = (const float*)d_in[8];
    const float* g2w2    = (const float*)d_in[9];
    const float* g2b2    = (const float*)d_in[10];
    const float* ln_g    = (const float*)d_in[11];
    const float* ln_b    = (const float*)d_in[12];
    const float* s2w1    = (const float*)d_in[13];
    const float* s2b1    = (const float*)d_in[14];
    const float* s2lng   = (const float*)d_in[15];
    const float* s2lnb   = (const float*)d_in[16];
    const float* s2w2    = (const float*)d_in[17];
    const float* s2b2    = (const float*)d_in[18];
    float* out = (float*)d_out;

    const int nNodes  = in_sizes[2];        // 200000
    const int nEdges  = in_sizes[1] / 2;    // 800000
    const int ATOM    = 128;
    const int HID     = 256;
    const int NG      = out_size / 2;       // 4096
    const int* src = ei;
    const int* dst = ei + nEdges;

    // Workspace layout (floats)
    float* buf0 = (float*)d_ws;                       // nNodes*256
    float* buf1 = buf0 + (size_t)nNodes * HID;        // nNodes*256
    float* buf2 = buf1 + (size_t)nNodes * HID;        // nNodes*256
    float* bufG = buf2 + (size_t)nNodes * HID;        // NG*256
    float* bufU = bufG + (size_t)NG * HID;            // NG*768

    const int TPB = 256;

    // 1. agg1 = x  (200000 x 128)
    {
        int n4 = nNodes * ATOM / 4;
        copy_f4_kernel<<<(n4 + TPB - 1) / TPB, TPB, 0, stream>>>(
            (const float4*)x, (float4*)buf0, n4);
    }
    // 2. agg1 += scatter x[src] -> dst
    {
        long threads = (long)nEdges * 32;
        scatter_add_kernel<<<(threads + TPB - 1) / TPB, TPB, 0, stream>>>(
            x, src, dst, buf0, nEdges, ATOM);
    }
    // 3. t1 = relu(agg1 @ g1w1 + g1b1)   [200000x128 @ 128x256]
    {
        int blocks = (nNodes / 64) * (HID / 64);
        wmma_gemm_kernel<<<blocks, 128, 0, stream>>>(
            buf0, g1w1, g1b1, buf1, nNodes, ATOM, HID, 1);
    }
    // 4. h1 = t1 @ g1w2 + g1b2           [200000x256 @ 256x256]
    {
        int blocks = (nNodes / 64) * (HID / 64);
        wmma_gemm_kernel<<<blocks, 128, 0, stream>>>(
            buf1, g1w2, g1b2, buf2, nNodes, HID, HID, 0);
    }
    // 5. h1 = leaky(LN(h1)), in place
    {
        long threads = (long)nNodes * 32;
        ln_leaky256_kernel<<<(threads + TPB - 1) / TPB, TPB, 0, stream>>>(
            buf2, ln_g, ln_b, nNodes);
    }
    // 6. agg2 = h1
    {
        int n4 = nNodes * HID / 4;
        copy_f4_kernel<<<(n4 + TPB - 1) / TPB, TPB, 0, stream>>>(
            (const float4*)buf2, (float4*)buf1, n4);
    }
    // 7. agg2 += scatter h1[src] -> dst
    {
        long threads = (long)nEdges * 32;
        scatter_add_kernel<<<(threads + TPB - 1) / TPB, TPB, 0, stream>>>(
            buf2, src, dst, buf1, nEdges, HID);
    }
    // 8. t2 = relu(agg2 @ g2w1 + g2b1)
    {
        int blocks = (nNodes / 64) * (HID / 64);
        wmma_gemm_kernel<<<blocks, 128, 0, stream>>>(
            buf1, g2w1, g2b1, buf0, nNodes, HID, HID, 1);
    }
    // 9. h2 = leaky(t2 @ g2w2 + g2b2, 0.2)
    {
        int blocks = (nNodes / 64) * (HID / 64);
        wmma_gemm_kernel<<<blocks, 128, 0, stream>>>(
            buf0, g2w2, g2b2, buf2, nNodes, HID, HID, 2);
    }
    // 10. g = -inf
    {
        int n = NG * HID;
        fill_neginf_kernel<<<(n + TPB - 1) / TPB, TPB, 0, stream>>>(bufG, n);
    }
    // 11. g = segment_max(h2, batch)
    {
        long threads = (long)nNodes * 32;
        seg_max_kernel<<<(threads + TPB - 1) / TPB, TPB, 0, stream>>>(
            buf2, batch, bufG, nNodes);
    }
    // 12. u = g @ s2w1 + s2b1            [4096x256 @ 256x768]
    {
        int blocks = (NG / 64) * (768 / 64);
        wmma_gemm_kernel<<<blocks, 128, 0, stream>>>(
            bufG, s2w1, s2b1, bufU, NG, HID, 768, 0);
    }
    // 13. out = softmax(leaky(LN(u)) @ s2w2 + s2b2)
    {
        long threads = (long)NG * 32;
        head_kernel<<<(threads + TPB - 1) / TPB, TPB, 0, stream>>>(
            bufU, s2lng, s2lnb, s2w2, s2b2, out, NG);
    }
}